// Attention_45165876085029
// MI455X (gfx1250) — compile-verified
//
#include <hip/hip_runtime.h>
#include <math.h>

// ---------------- problem constants ----------------
#define BB   256
#define TT   1024
#define ENC  512
#define RNN  1024
#define ADIM 128
#define NF   32
#define KW   31
#define PADW 15
#define RW   16
#define WIN  33   // 2R+1

typedef __attribute__((ext_vector_type(2))) float v2f;
typedef __attribute__((ext_vector_type(8))) float v8f;

// =====================================================================
// Kernel 1: query = hidden[256,1024] @ W_query^T[1024,128] via
//           V_WMMA_F32_16X16X4_F32.  One wave per 16x16 output tile.
// A layout (16x4 f32): lanes 0-15 -> M rows, K = k+0/k+1 in v0/v1;
//                      lanes16-31 -> same M rows, K = k+2/k+3.
// B layout (4x16 f32): lanes 0-15 -> N cols, K rows k+0/k+1;
//                      lanes16-31 -> K rows k+2/k+3.   B[k][n] = Wq[n][k].
// =====================================================================
__global__ __launch_bounds__(128) void query_gemm_wmma(
    const float* __restrict__ A,    // [256,1024]
    const float* __restrict__ Wq,   // [128,1024]
    float* __restrict__ Q)          // [256,128]
{
    const int lane  = threadIdx.x & 31;
    const int wave  = blockIdx.x * 4 + (threadIdx.x >> 5); // 0..127
    const int mTile = wave >> 3;                           // 0..15
    const int nTile = wave & 7;                            // 0..7

    const int mRow = mTile * 16 + (lane & 15);
    const int nCol = nTile * 16 + (lane & 15);
    const int koff = (lane >> 4) * 2;                      // 0 or 2

    const float* __restrict__ aRow = A  + (size_t)mRow * RNN + koff;
    const float* __restrict__ bRow = Wq + (size_t)nCol * RNN + koff;

    v8f c = {};
#pragma unroll 8
    for (int k = 0; k < RNN; k += 4) {
        v2f a = *(const v2f*)(aRow + k);
        v2f b = *(const v2f*)(bRow + k);
        c = __builtin_amdgcn_wmma_f32_16x16x4_f32(
                /*neg_a=*/false, a, /*neg_b=*/false, b,
                /*c_mod=*/(short)0, c, /*reuse_a=*/false, /*reuse_b=*/false);
    }

    // C/D layout: VGPR r -> M = r (lanes 0-15) / M = r+8 (lanes 16-31), N = lane&15
    const int rowBase = mTile * 16 + ((lane >> 4) << 3);
    const int col     = nTile * 16 + (lane & 15);
#pragma unroll
    for (int r = 0; r < 8; ++r)
        Q[(size_t)(rowBase + r) * ADIM + col] = c[r];
}

// =====================================================================
// Kernel 2: windowed attention. One block (256 thr = 8 wave32) per batch.
// Only the 33-position window contributes (everything else is -inf in the
// reference), so conv/dense/tanh/softmax/context are evaluated on 33 rows.
// =====================================================================
__global__ __launch_bounds__(256) void window_attention(
    const float* __restrict__ mem,    // [B,T,ENC]
    const float* __restrict__ pm,     // [B,T,ADIM]
    const float* __restrict__ awcat,  // [B,2,T]
    const float* __restrict__ cpos,   // [B]
    const int*   __restrict__ lens,   // [B]
    const float* __restrict__ convw,  // [NF,2,KW]
    const float* __restrict__ Wd,     // [ADIM,NF]
    const float* __restrict__ vw,     // [1,ADIM]
    const float* __restrict__ wpo,    // [1]
    const float* __restrict__ Q,      // [B,ADIM] from kernel 1
    float* __restrict__ outCtx,       // [B,ENC]
    float* __restrict__ outW,         // [B,T]
    float* __restrict__ outPos)       // [B]
{
    __shared__ float sLoc[WIN * NF];       // conv features at window positions
    __shared__ float sQ[ADIM];
    __shared__ float sV[ADIM];
    __shared__ float sWd[ADIM * NF];
    __shared__ float sCw[NF * 2 * KW];
    __shared__ float sAlign[WIN];
    __shared__ float sWgt[WIN];

    const int b   = blockIdx.x;
    const int tid = threadIdx.x;
    const int len = lens[b];

    // ---- window start (matches reference: clamp then round-to-even) ----
    float cp = cpos[b] + wpo[0];
    cp = fminf(fmaxf(cp, (float)RW), (float)(len - 1 - RW));
    const int ws = (int)rintf(fmaxf(cp - (float)RW, 0.0f));

    // ---- stage A: stage small operands in LDS ----
    for (int i = tid; i < ADIM; i += 256) { sQ[i] = Q[(size_t)b * ADIM + i]; sV[i] = vw[i]; }
    for (int i = tid; i < ADIM * NF; i += 256) sWd[i] = Wd[i];
    for (int i = tid; i < NF * 2 * KW; i += 256) sCw[i] = convw[i];
    __syncthreads();

    // ---- stage B: 1D conv, only at the 33 window positions ----
    for (int v = tid; v < WIN * NF; v += 256) {
        const int tt = v >> 5, f = v & 31;
        const int t = ws + tt;
        float acc = 0.0f;
#pragma unroll
        for (int c = 0; c < 2; ++c) {
            const float* __restrict__ in = awcat + ((size_t)b * 2 + c) * TT;
            const float* __restrict__ w  = sCw + (f * 2 + c) * KW;
            for (int k = 0; k < KW; ++k) {
                const int ti = t + k - PADW;
                if (ti >= 0 && ti < TT) acc += w[k] * in[ti];
            }
        }
        sLoc[v] = acc;
    }
    __syncthreads();

    // ---- stage C: alignment = v . tanh(Wd*loc + q + pm)  (wave per tt) ----
    const int lane = tid & 31, warp = tid >> 5;
    for (int tt = warp; tt < WIN; tt += 8) {
        const int t = ws + tt;
        const bool valid = (t < TT) && (t < len);      // wave-uniform
        float partial = 0.0f;
        if (valid) {
            const float* __restrict__ pmr = pm + ((size_t)b * TT + t) * ADIM;
            const float* __restrict__ lc  = sLoc + tt * NF;
#pragma unroll
            for (int j = 0; j < 4; ++j) {
                const int a = lane + 32 * j;
                float s = sQ[a] + pmr[a];
                const float* __restrict__ wd = sWd + a * NF;
#pragma unroll
                for (int f = 0; f < NF; ++f) s += wd[f] * lc[f];
                partial += sV[a] * tanhf(s);
            }
#pragma unroll
            for (int off = 16; off > 0; off >>= 1)
                partial += __shfl_xor(partial, off, 32);
        }
        if (lane == 0) sAlign[tt] = valid ? partial : -INFINITY;
    }
    __syncthreads();

    // ---- stage D: 33-wide softmax + expected position (trivial, serial) ----
    if (tid == 0) {
        float m = -INFINITY;
        for (int tt = 0; tt < WIN; ++tt) m = fmaxf(m, sAlign[tt]);
        float s = 0.0f;
        for (int tt = 0; tt < WIN; ++tt) { float e = expf(sAlign[tt] - m); sWgt[tt] = e; s += e; }
        const float inv = 1.0f / s;
        float np = 0.0f;
        for (int tt = 0; tt < WIN; ++tt) {
            const float w = sWgt[tt] * inv;
            sWgt[tt] = w;
            np += w * (float)(ws + tt);
        }
        outPos[b] = np;
    }
    __syncthreads();

    // ---- stage E: context over the 33 window rows, float2 per thread ----
    // 256 threads x 2 floats = 512 = ENC exactly; b64 loads (256B/wave-instr)
    {
        const float2* __restrict__ mrow0 = (const float2*)(mem + (size_t)b * TT * ENC);
        float2 acc; acc.x = 0.0f; acc.y = 0.0f;
        for (int tt = 0; tt < WIN; ++tt) {
            const int t = ws + tt;
            if (t < TT) {
                const float2 m2 = mrow0[(size_t)t * (ENC / 2) + tid];
                const float w = sWgt[tt];
                acc.x += w * m2.x;
                acc.y += w * m2.y;
            }
        }
        ((float2*)(outCtx + (size_t)b * ENC))[tid] = acc;
    }

    // ---- stage F: full attention-weight row, float4 per thread ----
    // 256 threads x 4 floats = 1024 = T exactly
    {
        const int t0 = tid * 4;
        float4 w4;
        {
            int d0 = t0 + 0 - ws; w4.x = (d0 >= 0 && d0 < WIN) ? sWgt[d0] : 0.0f;
            int d1 = t0 + 1 - ws; w4.y = (d1 >= 0 && d1 < WIN) ? sWgt[d1] : 0.0f;
            int d2 = t0 + 2 - ws; w4.z = (d2 >= 0 && d2 < WIN) ? sWgt[d2] : 0.0f;
            int d3 = t0 + 3 - ws; w4.w = (d3 >= 0 && d3 < WIN) ? sWgt[d3] : 0.0f;
        }
        ((float4*)(outW + (size_t)b * TT))[tid] = w4;
    }
}

// =====================================================================
extern "C" void kernel_launch(void* const* d_in, const int* in_sizes, int n_in,
                              void* d_out, int out_size, void* d_ws, size_t ws_size,
                              hipStream_t stream) {
    const float* hidden = (const float*)d_in[0];   // [B,RNN]
    const float* mem    = (const float*)d_in[1];   // [B,T,ENC]
    const float* pm     = (const float*)d_in[2];   // [B,T,ADIM]
    const float* awcat  = (const float*)d_in[3];   // [B,2,T]
    const float* cpos   = (const float*)d_in[4];   // [B]
    /* d_in[5] = mask: derived from memory_lengths, unused */
    const int*   lens   = (const int*)d_in[6];     // [B]
    const float* Wq     = (const float*)d_in[7];   // [ADIM,RNN]
    const float* convw  = (const float*)d_in[8];   // [NF,2,KW]
    const float* Wd     = (const float*)d_in[9];   // [ADIM,NF]
    const float* vw     = (const float*)d_in[10];  // [1,ADIM]
    const float* wpo    = (const float*)d_in[11];  // [1]

    float* Qws = (float*)d_ws;                     // [B,ADIM] scratch

    float* out    = (float*)d_out;
    float* outCtx = out;                           // [B,ENC]
    float* outW   = out + (size_t)BB * ENC;        // [B,T]
    float* outPos = outW + (size_t)BB * TT;        // [B]

    // 128 wave-tiles (16 M-tiles x 8 N-tiles), 4 waves per block
    query_gemm_wmma<<<32, 128, 0, stream>>>(hidden, Wq, Qws);
    window_attention<<<BB, 256, 0, stream>>>(mem, pm, awcat, cpos, lens,
                                             convw, Wd, vw, wpo, Qws,
                                             outCtx, outW, outPos);
}